// GIN_59854664237550
// MI455X (gfx1250) — compile-verified
//
#include <hip/hip_runtime.h>
#include <hip/hip_bf16.h>

// ---------------------------------------------------------------------------
// GIN forward on gfx1250 (MI455X).  fp32 WMMA (V_WMMA_F32_16X16X4_F32) for the
// 50000x128 @ 128x128 GEMMs; L2-resident atomic scatter-adds for the graph
// aggregation and pooling.
// ---------------------------------------------------------------------------

typedef float v2f __attribute__((ext_vector_type(2)));
typedef float v8f __attribute__((ext_vector_type(8)));

#define N_NODES 50000
#define N_EDGES 800000
#define NUM_GRAPHS 512
#define HID 128
#define OUTC 64
#define BN_EPS 1e-5f

// ---------------------------------------------------------------------- utils
__global__ void zero_buf(float* __restrict__ p, int n) {
    int i = blockIdx.x * blockDim.x + threadIdx.x;
    if (i < n) p[i] = 0.0f;
}

// agg[i] = (1 + eps[layer]) * x[i]
__global__ void init_agg(const float* __restrict__ X, float* __restrict__ AGG,
                         const float* __restrict__ epsc, int layer, int n) {
    int i = blockIdx.x * blockDim.x + threadIdx.x;
    if (i < n) AGG[i] = (1.0f + epsc[layer]) * X[i];
}

// one wave per edge; each lane handles 4 channels. agg[dst] += x[src]
__global__ void scatter_add(const float* __restrict__ X, const int* __restrict__ src,
                            const int* __restrict__ dst, float* __restrict__ AGG,
                            int nEdges) {
    int gid  = blockIdx.x * blockDim.x + threadIdx.x;
    int e    = gid >> 5;
    int lane = gid & 31;
    if (e >= nEdges) return;
    int s = src[e];
    int d = dst[e];
    const float* xr = X + (size_t)s * HID;
    float* ar = AGG + (size_t)d * HID;
    int c = lane * 4;
    float4 v = *(const float4*)(xr + c);
    atomicAdd(ar + c + 0, v.x);
    atomicAdd(ar + c + 1, v.y);
    atomicAdd(ar + c + 2, v.z);
    atomicAdd(ar + c + 3, v.w);
}

// ------------------------------------------------------------ WMMA fp32 GEMM
// Y[M x 128] = relu(X[M x 128] @ W[128 x 128] + bias)
// 4 waves / block; each wave computes one 16-row strip across all 128 cols
// (8 accumulators), iterating K in steps of 4 with V_WMMA_F32_16X16X4_F32.
//
// A (16x4 f32, 2 VGPRs): lane L -> row M=L%16, K = 2*(L/16) + {0,1}
// B (4x16  f32, 2 VGPRs): lane L -> col N=L%16, K = 2*(L/16) + {0,1}
// C/D (16x16 f32, 8 VGPRs): lane L, vgpr v -> (M = v + 8*(L/16), N = L%16)
__global__ void gemm_relu_wmma(const float* __restrict__ X,
                               const float* __restrict__ W,
                               const float* __restrict__ bias,
                               float* __restrict__ Y, int M) {
    int wave = threadIdx.x >> 5;            // 0..3
    int lane = threadIdx.x & 31;
    int tile = blockIdx.x * 4 + wave;       // 16-row tile index
    if (tile * 16 >= M) return;             // wave-uniform guard (EXEC all-1s)
    int m0 = tile * 16;
    int r  = lane & 15;
    int hi = lane >> 4;

    v8f acc[8] = {};                        // 8 n-tiles of 16 cols

    const float* xrow = X + (size_t)(m0 + r) * HID;
#pragma unroll 4
    for (int k = 0; k < HID; k += 4) {
        v2f a;
        a.x = xrow[k + 2 * hi];
        a.y = xrow[k + 2 * hi + 1];
        const float* wr0 = W + (size_t)(k + 2 * hi) * HID + r;
        const float* wr1 = wr0 + HID;
#pragma unroll
        for (int t = 0; t < 8; ++t) {
            v2f b;
            b.x = wr0[t * 16];
            b.y = wr1[t * 16];
            acc[t] = __builtin_amdgcn_wmma_f32_16x16x4_f32(
                false, a, false, b, (short)0, acc[t], false, false);
        }
    }

#pragma unroll
    for (int t = 0; t < 8; ++t) {
        int n = t * 16 + r;
        float bn = bias[n];
#pragma unroll
        for (int v = 0; v < 8; ++v) {
            int m = m0 + v + 8 * hi;
            float val = acc[t][v] + bn;
            Y[(size_t)m * HID + n] = fmaxf(val, 0.0f);
        }
    }
}

// ------------------------------------------------------------------ BatchNorm
// per-channel sum / sum-of-squares -> stats[0:128], stats[128:256]
__global__ void bn_stats(const float* __restrict__ H, float* __restrict__ stats,
                         int nRows, int rowsPerBlock) {
    int c  = threadIdx.x;                   // 128 threads = channels
    int r0 = blockIdx.x * rowsPerBlock;
    int r1 = min(r0 + rowsPerBlock, nRows);
    float s = 0.0f, s2 = 0.0f;
    for (int r = r0; r < r1; ++r) {
        float v = H[(size_t)r * HID + c];
        s += v;
        s2 += v * v;
    }
    atomicAdd(&stats[c], s);
    atomicAdd(&stats[HID + c], s2);
}

__global__ void bn_apply(const float* __restrict__ H, const float* __restrict__ stats,
                         const float* __restrict__ gamma, const float* __restrict__ beta,
                         float* __restrict__ Xn, int nRows) {
    int i = blockIdx.x * blockDim.x + threadIdx.x;
    if (i >= nRows * HID) return;
    int c = i & (HID - 1);
    float inv_n = 1.0f / (float)nRows;
    float mu  = stats[c] * inv_n;
    float var = stats[HID + c] * inv_n - mu * mu;
    float inv = rsqrtf(var + BN_EPS);
    Xn[i] = gamma[c] * (H[i] - mu) * inv + beta[c];
}

// --------------------------------------------------------------------- pooling
__global__ void pool_scatter(const float* __restrict__ X, const int* __restrict__ batch,
                             float* __restrict__ sums, float* __restrict__ cnts,
                             int nNodes) {
    int i = blockIdx.x * blockDim.x + threadIdx.x;
    if (i >= nNodes * HID) return;
    int node = i >> 7;
    int c = i & (HID - 1);
    int g = batch[node];
    atomicAdd(&sums[(size_t)g * HID + c], X[i]);
    if (c == 0) atomicAdd(&cnts[g], 1.0f);
}

// ------------------------------------------- final MLP + log_softmax per graph
__global__ void final_mlp(const float* __restrict__ sums, const float* __restrict__ cnts,
                          const float* __restrict__ w1, const float* __restrict__ b1,
                          const float* __restrict__ w2, const float* __restrict__ b2,
                          float* __restrict__ out) {
    __shared__ float p[HID];
    __shared__ float h[HID];
    __shared__ float lg[OUTC];
    __shared__ float red[2];
    int g = blockIdx.x;
    int t = threadIdx.x;                    // 128 threads

    float cnt = fmaxf(cnts[g], 1.0f);
    p[t] = sums[(size_t)g * HID + t] / cnt;
    __syncthreads();

    float acc = b1[t];
    for (int k = 0; k < HID; ++k) acc = fmaf(p[k], w1[(size_t)k * HID + t], acc);
    h[t] = fmaxf(acc, 0.0f);
    __syncthreads();

    if (t < OUTC) {
        float a2 = b2[t];
        for (int k = 0; k < HID; ++k) a2 = fmaf(h[k], w2[(size_t)k * OUTC + t], a2);
        lg[t] = a2;
    }
    __syncthreads();

    if (t == 0) {
        float m = lg[0];
        for (int o = 1; o < OUTC; ++o) m = fmaxf(m, lg[o]);
        float se = 0.0f;
        for (int o = 0; o < OUTC; ++o) se += expf(lg[o] - m);
        red[0] = m;
        red[1] = logf(se);
    }
    __syncthreads();

    if (t < OUTC) out[(size_t)g * OUTC + t] = lg[t] - red[0] - red[1];
}

// ---------------------------------------------------------------------- launch
extern "C" void kernel_launch(void* const* d_in, const int* in_sizes, int n_in,
                              void* d_out, int out_size, void* d_ws, size_t ws_size,
                              hipStream_t stream) {
    const float* x     = (const float*)d_in[0];
    const int*   ei    = (const int*)d_in[1];
    const int*   batch = (const int*)d_in[2];
    const float* W1    = (const float*)d_in[3];
    const float* b1    = (const float*)d_in[4];
    const float* W2    = (const float*)d_in[5];
    const float* b2    = (const float*)d_in[6];
    const float* gamma = (const float*)d_in[7];
    const float* beta  = (const float*)d_in[8];
    const float* epsc  = (const float*)d_in[9];
    const float* l1w   = (const float*)d_in[10];
    const float* l1b   = (const float*)d_in[11];
    const float* l2w   = (const float*)d_in[12];
    const float* l2b   = (const float*)d_in[13];

    const size_t NF = (size_t)N_NODES * HID;        // 6.4M floats
    float* ws    = (float*)d_ws;
    float* B0    = ws;                              // x_cur / x_next
    float* B1    = ws + NF;                         // agg / t2
    float* B2    = ws + 2 * NF;                     // t1
    float* stats = ws + 3 * NF;                     // 256 floats
    float* psums = stats + 256;                     // 512*128
    float* pcnts = psums + (size_t)NUM_GRAPHS * HID; // 512

    const int* src = ei;
    const int* dst = ei + N_EDGES;

    const int nElem       = (int)NF;
    const int elemBlocks  = (nElem + 255) / 256;
    const int edgeBlocks  = (N_EDGES * 32) / 256;   // one wave per edge
    const int gemmBlocks  = (N_NODES / 16 + 3) / 4; // 4 waves/block, 1 tile/wave
    const int rowsPerBlk  = 256;
    const int statBlocks  = (N_NODES + rowsPerBlk - 1) / rowsPerBlk;

    for (int l = 0; l < 3; ++l) {
        const float* xs = (l == 0) ? x : B0;

        // agg = (1+eps)*x ; then agg[dst] += x[src]
        init_agg<<<elemBlocks, 256, 0, stream>>>(xs, B1, epsc, l, nElem);
        scatter_add<<<edgeBlocks, 256, 0, stream>>>(xs, src, dst, B1, N_EDGES);

        // t1 = relu(agg @ W1 + b1) ; t2 = relu(t1 @ W2 + b2)
        gemm_relu_wmma<<<gemmBlocks, 128, 0, stream>>>(B1, W1 + (size_t)l * HID * HID,
                                                       b1 + l * HID, B2, N_NODES);
        gemm_relu_wmma<<<gemmBlocks, 128, 0, stream>>>(B2, W2 + (size_t)l * HID * HID,
                                                       b2 + l * HID, B1, N_NODES);

        // batchnorm (training-mode batch stats, biased variance)
        zero_buf<<<1, 256, 0, stream>>>(stats, 256);
        bn_stats<<<statBlocks, 128, 0, stream>>>(B1, stats, N_NODES, rowsPerBlk);
        bn_apply<<<elemBlocks, 256, 0, stream>>>(B1, stats, gamma + l * HID,
                                                 beta + l * HID, B0, N_NODES);
    }

    // global mean pool + MLP head + log_softmax
    const int poolZero = NUM_GRAPHS * HID + NUM_GRAPHS;
    zero_buf<<<(poolZero + 255) / 256, 256, 0, stream>>>(psums, poolZero);
    pool_scatter<<<elemBlocks, 256, 0, stream>>>(B0, batch, psums, pcnts, N_NODES);
    final_mlp<<<NUM_GRAPHS, 128, 0, stream>>>(psums, pcnts, l1w, l1b, l2w, l2b,
                                              (float*)d_out);
}